// OADIS_51719996178963
// MI455X (gfx1250) — compile-verified
//
#include <hip/hip_runtime.h>
#include <hip/hip_bf16.h>
#include <math.h>

typedef __attribute__((ext_vector_type(2))) float v2f;
typedef __attribute__((ext_vector_type(8))) float v8f;

typedef __attribute__((address_space(1))) int glb_i32_t;
typedef __attribute__((address_space(3))) int lds_i32_t;

#define BS   256
#define DD   1024
#define LL   49
#define LP   64
#define KT   32
#define KP   (KT + 2)   /* 34-dword row pitch: conflict-free, 8B-aligned pairs */
#define LSTR 65
#define EMB  300
#define NA   400
#define NO   500
#define LAMB 10.0f
#define TEMP 0.05f
#define EPSR 1e-8f

// workspace layout (float offsets)
#define WS_VECS 0                              // 2*3*256*64   = 98304
#define WS_FEAT (WS_VECS + 2*3*BS*LP)          // 6*256*1024   = 1572864
#define WS_AW   (WS_FEAT + 6*BS*DD)            // 400*300
#define WS_OW   (WS_AW + NA*EMB)               // 500*300
#define WS_AWN  (WS_OW + NO*EMB)               // 400
#define WS_OWN  (WS_AWN + NA)                  // 500
#define WS_ACC  (WS_OWN + NO)                  // 8

// ---- gfx1250 async global->LDS copy (ASYNCcnt-tracked) ---------------------
__device__ __forceinline__ void async_ld32(const float* gsrc, float* ldst) {
#if __has_builtin(__builtin_amdgcn_global_load_async_to_lds_b32)
  __builtin_amdgcn_global_load_async_to_lds_b32(
      (glb_i32_t*)gsrc, (lds_i32_t*)ldst, 0, 0);
#else
  unsigned int lo = (unsigned int)(unsigned long long)(void*)ldst; // low 32b = LDS offset
  unsigned long long ga = (unsigned long long)gsrc;
  asm volatile("global_load_async_to_lds_b32 %0, %1, off"
               :: "v"(lo), "v"(ga) : "memory");
#endif
}
__device__ __forceinline__ void wait_async0() {
#if __has_builtin(__builtin_amdgcn_s_wait_asynccnt)
  __builtin_amdgcn_s_wait_asynccnt(0);
#else
  asm volatile("s_wait_asynccnt 0" ::: "memory");
#endif
}

// ---------------------------------------------------------------- stage A ---
// Per (batch, pair): Gram(49x49, K=1024) via v_wmma_f32_16x16x4_f32 with
// async global->LDS tile fill, rank-1 normalization, then row/col/inverse
// softmax reductions + renorm.
__global__ __launch_bounds__(128)
void stageA(const float* __restrict__ img1, const float* __restrict__ img2a,
            const float* __restrict__ img2o, float* __restrict__ ws) {
  __shared__ float sA[LP][KP];      // transposed tile: [l][k]
  __shared__ float sB[LP][KP];
  __shared__ float rel[LP][LSTR];
  __shared__ float nrm1s[LP], nrm2s[LP];
  __shared__ float rmax[LP], rsumP[LP], rmin[LP], rsumN[LP], cmx[LP], csm[LP];
  __shared__ float outv[3][LP];
  __shared__ float tots[3];

  const int b = blockIdx.x;
  const int pair = blockIdx.y;
  const float* __restrict__ A = img1 + (size_t)b * DD * LL;
  const float* __restrict__ B = (pair == 0 ? img2a : img2o) + (size_t)b * DD * LL;

  const int t = threadIdx.x;
  const int lane = t & 31, wave = t >> 5;
  const int m = lane & 15;
  const int kb = (lane >> 4) * 2;

  // pad norms default to 1 (pad rows of rel are never read anyway)
  if (t < LP) { nrm1s[t] = 1.f; nrm2s[t] = 1.f; }
  // zero the pad rows l in [49,64) once; async fills never touch them
  for (int idx = t; idx < (LP - LL) * KP; idx += 128) {
    const int r = idx / KP, c = idx % KP;
    sA[LL + r][c] = 0.f;
    sB[LL + r][c] = 0.f;
  }

  float nacc = 0.f;     // per-thread column-norm accumulator
  v8f acc[4] = {};

  for (int k0 = 0; k0 < DD; k0 += KT) {
    // async DMA-style tile fill: global (d-major, coalesced) -> LDS [l][k]
    for (int idx = t; idx < KT * LL; idx += 128) {
      const int dd = idx / LL;
      const int l  = idx - dd * LL;
      async_ld32(A + (size_t)(k0 + dd) * LL + l, &sA[l][dd]);
      async_ld32(B + (size_t)(k0 + dd) * LL + l, &sB[l][dd]);
    }
    wait_async0();
    __syncthreads();

    // column norms: contiguous per-thread runs over the tile
    if (t < LL) {
      for (int dd = 0; dd < KT; dd++) { const float v = sA[t][dd]; nacc += v * v; }
    } else if (t >= 64 && t < 64 + LL) {
      for (int dd = 0; dd < KT; dd++) { const float v = sB[t - 64][dd]; nacc += v * v; }
    }

#pragma unroll
    for (int kk = 0; kk < KT; kk += 4) {
      const v2f av = *(const v2f*)__builtin_assume_aligned(&sA[wave * 16 + m][kk + kb], 8);
#pragma unroll
      for (int nt = 0; nt < 4; nt++) {
        const v2f bv = *(const v2f*)__builtin_assume_aligned(&sB[nt * 16 + m][kk + kb], 8);
        acc[nt] = __builtin_amdgcn_wmma_f32_16x16x4_f32(
            false, av, false, bv, (short)0, acc[nt], false, false);
      }
    }
    __syncthreads();
  }

  if (t < LL)                        nrm1s[t]      = fmaxf(sqrtf(nacc), 1e-12f);
  else if (t >= 64 && t < 64 + LL)   nrm2s[t - 64] = fmaxf(sqrtf(nacc), 1e-12f);
  __syncthreads();

  // scatter accumulators into rel, applying the rank-1 normalization
#pragma unroll
  for (int nt = 0; nt < 4; nt++) {
    const int col = nt * 16 + m;
#pragma unroll
    for (int r = 0; r < 8; r++) {
      const int row = wave * 16 + r + (lane >> 4) * 8;
      rel[row][col] = acc[nt][r] / (nrm1s[row] * nrm2s[col]);
    }
  }
  __syncthreads();

  if (t < LL) {
    float mx = -1e30f, mn = 1e30f;
    for (int j = 0; j < LL; j++) { float v = rel[t][j]; mx = fmaxf(mx, v); mn = fminf(mn, v); }
    float sp = 0.f, sn = 0.f;
    for (int j = 0; j < LL; j++) {
      float v = rel[t][j];
      sp += expf((v - mx) * LAMB);
      sn += expf((mn - v) * LAMB);
    }
    rmax[t] = mx; rmin[t] = mn; rsumP[t] = sp; rsumN[t] = sn;
    float cm = -1e30f;
    for (int i = 0; i < LL; i++) cm = fmaxf(cm, rel[i][t]);
    float cs = 0.f;
    for (int i = 0; i < LL; i++) cs += expf((rel[i][t] - cm) * LAMB);
    cmx[t] = cm; csm[t] = cs;
  }
  __syncthreads();

  if (t < LL) {
    float s12 = 0.f, d12 = 0.f, s21 = 0.f;
    for (int i = 0; i < LL; i++) {
      const float v = rel[i][t];
      s12 += expf((v - rmax[i]) * LAMB) / rsumP[i];
      d12 += expf((rmin[i] - v) * LAMB) / rsumN[i];
    }
    for (int j = 0; j < LL; j++)
      s21 += expf((rel[t][j] - cmx[j]) * LAMB) / csm[j];
    outv[0][t] = s12; outv[1][t] = s21; outv[2][t] = d12;
  }
  __syncthreads();
  if (t < 3) { float s = 0.f; for (int l = 0; l < LL; l++) s += outv[t][l]; tots[t] = s; }
  __syncthreads();

  float* vecs = ws + WS_VECS;
  for (int idx = t; idx < 3 * LP; idx += 128) {
    const int ty = idx >> 6, l = idx & (LP - 1);
    const float v = (l < LL) ? outv[ty][l] / (tots[ty] + EPSR) : 0.f;
    vecs[(((size_t)pair * 3 + ty) * BS + b) * LP + l] = v;
  }
}

// ---------------------------------------------------------------- stage B ---
// feat[b,d] = sum_l img[b,d,l] * w[b,l]  (two weight vectors per image pass)
__global__ __launch_bounds__(256)
void stageB(const float* __restrict__ img1, const float* __restrict__ img2a,
            const float* __restrict__ img2o, float* __restrict__ ws) {
  const int b = blockIdx.x, im = blockIdx.y, t = threadIdx.x;
  const float* vecs = ws + WS_VECS;
  float* feat = ws + WS_FEAT;
  const float* src; const float *w0p, *w1p; int f0, f1;
  if (im == 0)      { src = img1;  w0p = &vecs[((0*3+1)*BS + b)*LP]; w1p = &vecs[((1*3+1)*BS + b)*LP]; f0 = 0; f1 = 3; }
  else if (im == 1) { src = img2a; w0p = &vecs[((0*3+0)*BS + b)*LP]; w1p = &vecs[((0*3+2)*BS + b)*LP]; f0 = 1; f1 = 2; }
  else              { src = img2o; w0p = &vecs[((1*3+0)*BS + b)*LP]; w1p = &vecs[((1*3+2)*BS + b)*LP]; f0 = 4; f1 = 5; }
  __shared__ float w0[LL], w1[LL];
  if (t < LL) { w0[t] = w0p[t]; w1[t] = w1p[t]; }
  __syncthreads();
  for (int d = t; d < DD; d += 256) {
    const float* row = src + ((size_t)b * DD + d) * LL;
    float s0 = 0.f, s1 = 0.f;
    for (int l = 0; l < LL; l++) { const float v = row[l]; s0 += v * w0[l]; s1 += v * w1[l]; }
    feat[(size_t)f0 * BS * DD + (size_t)b * DD + d] = s0;
    feat[(size_t)f1 * BS * DD + (size_t)b * DD + d] = s1;
  }
}

// ---------------------------------------------------------------- stage C1 --
// Six (256x1024)@(1024x300)+bias GEMMs via WMMA, written directly to d_out.
__global__ __launch_bounds__(128)
void stageC1(const float* __restrict__ ws, const float* __restrict__ Wattr,
             const float* __restrict__ battr, const float* __restrict__ Wobj,
             const float* __restrict__ bobj, float* __restrict__ out) {
  const int t = threadIdx.x, lane = t & 31, wave = t >> 5;
  const int wg = blockIdx.x * 4 + wave;
  const int NT = 19, MT = 16;
  if (wg >= 6 * MT * NT) return;
  const int mat = wg / (MT * NT);
  const int rem = wg % (MT * NT);
  const int mt = rem / NT, nt = rem % NT;
  const int fidx[6] = {5, 2, 0, 1, 3, 4};
  const bool useAttr = (mat == 0) || (mat == 2) || (mat == 3);
  const float* A = ws + WS_FEAT + (size_t)fidx[mat] * BS * DD;
  const float* W = useAttr ? Wattr : Wobj;
  const float* bias = useAttr ? battr : bobj;
  float* o = out + 2 + (size_t)mat * BS * EMB;

  const int m = lane & 15, kb = (lane >> 4) * 2;
  const int row = mt * 16 + m;
  const int n = nt * 16 + m;
  const bool nok = n < EMB;
  v8f acc = {};
  for (int k0 = 0; k0 < DD; k0 += 4) {
    v2f av;
    av.x = A[(size_t)row * DD + k0 + kb];
    av.y = A[(size_t)row * DD + k0 + kb + 1];
    v2f bv;
    bv.x = nok ? W[(size_t)(k0 + kb) * EMB + n] : 0.f;
    bv.y = nok ? W[(size_t)(k0 + kb + 1) * EMB + n] : 0.f;
    acc = __builtin_amdgcn_wmma_f32_16x16x4_f32(false, av, false, bv, (short)0, acc, false, false);
  }
  if (nok) {
    const float bb = bias[n];
    const int rbase = mt * 16 + (lane >> 4) * 8;
#pragma unroll
    for (int r = 0; r < 8; r++)
      o[(size_t)(rbase + r) * EMB + n] = acc[r] + bb;
  }
}

// ---------------------------------------------------------------- stage C2 --
// attr_weight = attr_table @ W_attr_mlp + b ; obj_weight likewise (WMMA).
__global__ __launch_bounds__(128)
void stageC2(const float* __restrict__ atab, const float* __restrict__ otab,
             const float* __restrict__ Wam, const float* __restrict__ bam,
             const float* __restrict__ Wom, const float* __restrict__ bom,
             float* __restrict__ ws) {
  const int t = threadIdx.x, lane = t & 31, wave = t >> 5;
  const int wg = blockIdx.x * 4 + wave;
  const int NT = 19;
  const int total0 = 25 * NT;            // attr: 400 rows -> 25 M-tiles
  const int total = total0 + 32 * NT;    // obj: 500 rows -> 32 M-tiles
  if (wg >= total) return;
  const int mat = (wg < total0) ? 0 : 1;
  const int rem = (mat == 0) ? wg : wg - total0;
  const int mt = rem / NT, nt = rem % NT;
  const int M = (mat == 0) ? NA : NO;
  const float* A = (mat == 0) ? atab : otab;
  const float* W = (mat == 0) ? Wam : Wom;
  const float* bias = (mat == 0) ? bam : bom;
  float* o = ws + ((mat == 0) ? WS_AW : WS_OW);

  const int m = lane & 15, kb = (lane >> 4) * 2;
  const int row = mt * 16 + m;
  const bool rok = row < M;
  const int n = nt * 16 + m;
  const bool nok = n < EMB;
  v8f acc = {};
  for (int k0 = 0; k0 < EMB; k0 += 4) {
    v2f av;
    av.x = rok ? A[(size_t)row * EMB + k0 + kb] : 0.f;
    av.y = rok ? A[(size_t)row * EMB + k0 + kb + 1] : 0.f;
    v2f bv;
    bv.x = nok ? W[(size_t)(k0 + kb) * EMB + n] : 0.f;
    bv.y = nok ? W[(size_t)(k0 + kb + 1) * EMB + n] : 0.f;
    acc = __builtin_amdgcn_wmma_f32_16x16x4_f32(false, av, false, bv, (short)0, acc, false, false);
  }
  if (nok) {
    const float bb = bias[n];
    const int rbase = mt * 16 + (lane >> 4) * 8;
#pragma unroll
    for (int r = 0; r < 8; r++) {
      const int rr = rbase + r;
      if (rr < M) o[(size_t)rr * EMB + n] = acc[r] + bb;
    }
  }
}

// row norms of the class-weight matrices
__global__ void normsK(float* __restrict__ ws) {
  const int c = blockIdx.x * blockDim.x + threadIdx.x;
  if (c >= NA + NO) return;
  const float* row;
  float* outn;
  if (c < NA) { row = ws + WS_AW + (size_t)c * EMB; outn = ws + WS_AWN + c; }
  else        { row = ws + WS_OW + (size_t)(c - NA) * EMB; outn = ws + WS_OWN + (c - NA); }
  float s = 0.f;
  for (int k = 0; k < EMB; k++) s += row[k] * row[k];
  *outn = fmaxf(sqrtf(s), 1e-12f);
}

// ---------------------------------------------------------------- stage C3 --
// cosine classify + masked CE accumulation
__global__ __launch_bounds__(128)
void stageC3(const float* __restrict__ out, const int* __restrict__ attr1,
             const int* __restrict__ obj1, const int* __restrict__ mask_task,
             float* __restrict__ ws) {
  const int b = blockIdx.x, term = blockIdx.y, t = threadIdx.x;
  if (mask_task[b] != 1) return;
  const bool isAttr = term < 2;
  const int NC = isAttr ? NA : NO;
  const float* W = ws + (isAttr ? WS_AW : WS_OW);
  const float* wn = ws + (isAttr ? WS_AWN : WS_OWN);
  const int label = isAttr ? attr1[b] : obj1[b];
  const float* feat = out + 2 + (size_t)(term + 2) * BS * EMB + (size_t)b * EMB;

  __shared__ float sf[EMB];
  __shared__ float logits[512];
  __shared__ float red[128];
  for (int k = t; k < EMB; k += 128) sf[k] = feat[k];
  __syncthreads();
  float p = 0.f;
  for (int k = t; k < EMB; k += 128) p += sf[k] * sf[k];
  red[t] = p;
  __syncthreads();
  for (int s = 64; s > 0; s >>= 1) {
    if (t < s) red[t] += red[t + s];
    __syncthreads();
  }
  const float fn = fmaxf(sqrtf(red[0]), 1e-12f);
  for (int c = t; c < NC; c += 128) {
    const float* wr = W + (size_t)c * EMB;
    float d = 0.f;
    for (int k = 0; k < EMB; k++) d += sf[k] * wr[k];
    logits[c] = d / (fn * wn[c] * TEMP);
  }
  __syncthreads();
  if (t == 0) {
    float mx = -1e30f;
    for (int c = 0; c < NC; c++) mx = fmaxf(mx, logits[c]);
    float se = 0.f;
    for (int c = 0; c < NC; c++) se += expf(logits[c] - mx);
    const float ce = (mx + logf(se)) - logits[label];
    atomicAdd(&ws[WS_ACC + term], ce);
  }
}

__global__ void initK(float* __restrict__ ws) {
  if (threadIdx.x < 8) ws[WS_ACC + threadIdx.x] = 0.f;
}

__global__ void finalK(const int* __restrict__ mask_task,
                       const float* __restrict__ ws, float* __restrict__ out) {
  if (threadIdx.x == 0) {
    int ms = 0;
    for (int b = 0; b < BS; b++) ms += (mask_task[b] == 1);
    const float den = fmaxf((float)ms, 1.f);
    out[0] = (ws[WS_ACC + 0] + ws[WS_ACC + 1]) * 0.5f / den;
    out[1] = (ws[WS_ACC + 2] + ws[WS_ACC + 3]) * 0.5f / den;
  }
}

extern "C" void kernel_launch(void* const* d_in, const int* in_sizes, int n_in,
                              void* d_out, int out_size, void* d_ws, size_t ws_size,
                              hipStream_t stream) {
  const float* img1   = (const float*)d_in[0];
  const float* img2a  = (const float*)d_in[1];
  const float* img2o  = (const float*)d_in[2];
  const int*   attr1  = (const int*)d_in[3];
  const int*   obj1   = (const int*)d_in[4];
  const int*   maskt  = (const int*)d_in[5];
  const float* Wsa    = (const float*)d_in[6];
  const float* bsa    = (const float*)d_in[7];
  const float* Wso    = (const float*)d_in[8];
  const float* bso    = (const float*)d_in[9];
  const float* atab   = (const float*)d_in[10];
  const float* otab   = (const float*)d_in[11];
  const float* Wam    = (const float*)d_in[12];
  const float* bam    = (const float*)d_in[13];
  const float* Wom    = (const float*)d_in[14];
  const float* bom    = (const float*)d_in[15];
  float* out = (float*)d_out;
  float* ws  = (float*)d_ws;

  initK<<<1, 32, 0, stream>>>(ws);
  stageA<<<dim3(BS, 2), 128, 0, stream>>>(img1, img2a, img2o, ws);
  stageB<<<dim3(BS, 3), 256, 0, stream>>>(img1, img2a, img2o, ws);
  stageC1<<<456, 128, 0, stream>>>(ws, Wsa, bsa, Wso, bso, out);
  stageC2<<<271, 128, 0, stream>>>(atab, otab, Wam, bam, Wom, bom, ws);
  normsK<<<8, 128, 0, stream>>>(ws);
  stageC3<<<dim3(BS, 4), 128, 0, stream>>>(out, attr1, obj1, maskt, ws);
  finalK<<<1, 32, 0, stream>>>(maskt, ws, out);
}